// DCK_1786706395589
// MI455X (gfx1250) — compile-verified
//
#include <hip/hip_runtime.h>
#include <hip/hip_bf16.h>
#include <math.h>

#define NN 4
#define CC 64
#define HH 64
#define WW 64
#define LL (HH * WW)        // 4096
#define LT (LL / 16)        // 256 l-tiles per sample
#define MCH 8               // m-chunks per (n, l-tile)
#define MT_PER_CH (LT / MCH)// 32 m-tiles per chunk
#define HB (HH / 4)         // 16 h-blocks for dwconv

typedef __attribute__((ext_vector_type(2))) float v2f;
typedef __attribute__((ext_vector_type(8))) float v8f;

// ---------------------------------------------------------------------------
// mul2 16x16 tile: C[l0:l0+16, m0:m0+16] = sum_c out2[n,c,l] * out3[n,c,m]
// A = out2^T tile (16x64), B = out3 tile (64x16); 16 chained f32 WMMAs (K=4).
// ---------------------------------------------------------------------------
__device__ __forceinline__ v8f mul2_tile(const float* __restrict__ out2,
                                         const float* __restrict__ out3,
                                         int n, int l0, int m0, int lane) {
  v8f c = {};
  const int half = lane >> 4;   // 0 or 1
  const int lm   = lane & 15;
  const float* o2 = out2 + (size_t)n * CC * LL + l0 + lm;  // A row = lm (l index)
  const float* o3 = out3 + (size_t)n * CC * LL + m0 + lm;  // B col = lm (m index)
#pragma unroll
  for (int kk = 0; kk < 16; ++kk) {
    const int kbase = kk * 4 + 2 * half;
    v2f a, b;
    a.x = o2[(size_t)(kbase + 0) * LL];
    a.y = o2[(size_t)(kbase + 1) * LL];
    b.x = o3[(size_t)(kbase + 0) * LL];
    b.y = o3[(size_t)(kbase + 1) * LL];
    c = __builtin_amdgcn_wmma_f32_16x16x4_f32(false, a, false, b,
                                              (short)0, c, false, false);
  }
  return c;
}

// ---------------------------------------------------------------------------
// init: zero accumulators (must re-run every call for graph replay)
// ---------------------------------------------------------------------------
__global__ void k_init(float* __restrict__ mul3, float* __restrict__ ssum,
                       unsigned* __restrict__ smaxk,
                       float* __restrict__ bnsum, float* __restrict__ bnsq) {
  int i = blockIdx.x * blockDim.x + threadIdx.x;
  if (i < NN * LL * 9) mul3[i] = 0.f;
  if (i < NN) { ssum[i] = 0.f; smaxk[i] = 0x007FFFFFu; /* key(-inf) */ }
  if (i < 3 * CC) { bnsum[i] = 0.f; bnsq[i] = 0.f; }
}

// ---------------------------------------------------------------------------
// out1/out2/out3: 1x1 convs (einsum nchw,oc->nohw + bias)
// ---------------------------------------------------------------------------
__global__ void k_out123(const float* __restrict__ x,
                         const float* __restrict__ w1, const float* __restrict__ b1,
                         const float* __restrict__ w2, const float* __restrict__ b2,
                         const float* __restrict__ w3, const float* __restrict__ b3,
                         float* __restrict__ out1, float* __restrict__ out2,
                         float* __restrict__ out3) {
  int idx = blockIdx.x * blockDim.x + threadIdx.x;     // 3*N*C*L threads
  int which = idx / (NN * CC * LL);
  int rem   = idx % (NN * CC * LL);
  int n = rem / (CC * LL);
  int o = (rem / LL) % CC;
  int l = rem % LL;
  const float* w = (which == 0) ? w1 : (which == 1) ? w2 : w3;
  const float* b = (which == 0) ? b1 : (which == 1) ? b2 : b3;
  float* out     = (which == 0) ? out1 : (which == 1) ? out2 : out3;
  const float* xr = x + (size_t)n * CC * LL + l;
  const float* wr = w + o * CC;
  float acc = b[o];
#pragma unroll 8
  for (int c = 0; c < CC; ++c) acc = fmaf(wr[c], xr[(size_t)c * LL], acc);
  out[(size_t)(n * CC + o) * LL + l] = acc;
}

// ---------------------------------------------------------------------------
// out4 = _paired(x): per-pixel exp(-(c1+c2+c3)/27), shape (N, L, 9)
// ---------------------------------------------------------------------------
__global__ void k_paired(const float* __restrict__ x,
                         const float* __restrict__ pcw1, const float* __restrict__ pcb1,
                         const float* __restrict__ pcw2, const float* __restrict__ pcb2,
                         const float* __restrict__ pcw3, const float* __restrict__ pcb3,
                         float* __restrict__ out4) {
  int idx = blockIdx.x * blockDim.x + threadIdx.x;     // N*L threads
  int n = idx / LL, l = idx % LL;
  int h = l / WW, w = l % WW;
  float acc[3][3] = {};
  const float* xn = x + (size_t)n * CC * LL;
  for (int c = 0; c < CC; ++c) {
    const float* xc = xn + (size_t)c * LL;
    float ctr = xc[l];
    float p[3][3];
#pragma unroll
    for (int i = 0; i < 3; ++i) {
#pragma unroll
      for (int j = 0; j < 3; ++j) {
        int hh = h + i - 1, ww = w + j - 1;
        float q = ((unsigned)hh < HH && (unsigned)ww < WW) ? xc[hh * WW + ww] : 0.f;
        p[i][j] = q - ctr;
      }
    }
    float w1c = pcw1[c];
    const float* w2c = pcw2 + c * 4;
    const float* w3c = pcw3 + c * 9;
#pragma unroll
    for (int i = 0; i < 3; ++i) {
#pragma unroll
      for (int j = 0; j < 3; ++j) {
        float s = p[i][j] * w1c;                        // c1
#pragma unroll
        for (int di = 0; di < 2; ++di) {                // c2: pad (1,0,1,0), 2x2 VALID
#pragma unroll
          for (int dj = 0; dj < 2; ++dj) {
            int ii = i + di - 1, jj = j + dj - 1;
            if (ii >= 0 && jj >= 0) s = fmaf(p[ii][jj], w2c[di * 2 + dj], s);
          }
        }
#pragma unroll
        for (int u = 0; u < 3; ++u) {                   // c3: 3x3, pad 1
#pragma unroll
          for (int v = 0; v < 3; ++v) {
            int ii = i + u - 1, jj = j + v - 1;
            if (ii >= 0 && ii < 3 && jj >= 0 && jj < 3)
              s = fmaf(p[ii][jj], w3c[u * 3 + v], s);
          }
        }
        acc[i][j] += s;
      }
    }
  }
  float bias = pcb1[0] + pcb2[0] + pcb3[0];
#pragma unroll
  for (int i = 0; i < 3; ++i)
#pragma unroll
    for (int j = 0; j < 3; ++j) {
      float s = acc[i][j] + bias;
      out4[((size_t)n * LL + l) * 9 + i * 3 + j] = __expf(s * (-1.f / 27.f));
    }
}

// ---------------------------------------------------------------------------
// mul1[n,c,m] = sum_l out1[n,c,l] * coarse[n,m,l]
// ---------------------------------------------------------------------------
__global__ void k_mul1(const float* __restrict__ out1, const float* __restrict__ coarse,
                       float* __restrict__ mul1) {
  int idx = blockIdx.x * blockDim.x + threadIdx.x;     // N*C*C threads
  int n = idx / (CC * CC);
  int c = (idx / CC) % CC;
  int m = idx % CC;
  const float* a = out1 + (size_t)(n * CC + c) * LL;
  const float* b = coarse + (size_t)(n * CC + m) * LL;
  float acc = 0.f;
  for (int l = 0; l < LL; ++l) acc = fmaf(a[l], b[l], acc);
  mul1[idx] = acc;
}

// ---------------------------------------------------------------------------
// Pass 1: global max of mul2 per sample (WMMA tiles + uint-key atomicMax)
// ---------------------------------------------------------------------------
__global__ void k_softmax_max(const float* __restrict__ out2,
                              const float* __restrict__ out3,
                              unsigned* __restrict__ smaxk) {
  int tid  = threadIdx.x;
  int lane = tid & 31;
  int wg   = blockIdx.x * (blockDim.x >> 5) + (tid >> 5);
  int n    = wg / (LT * MCH);
  int rem  = wg % (LT * MCH);
  int lt   = rem / MCH;
  int mc   = rem % MCH;
  int l0   = lt * 16;
  float vmax = -3.4e38f;
  for (int mt = 0; mt < MT_PER_CH; ++mt) {
    int m0 = (mc * MT_PER_CH + mt) * 16;
    v8f c = mul2_tile(out2, out3, n, l0, m0, lane);
#pragma unroll
    for (int r = 0; r < 8; ++r) vmax = fmaxf(vmax, c[r]);
  }
#pragma unroll
  for (int off = 16; off > 0; off >>= 1) vmax = fmaxf(vmax, __shfl_xor(vmax, off, 32));
  if (lane == 0) {
    unsigned uv  = __float_as_uint(vmax);
    unsigned key = (uv & 0x80000000u) ? ~uv : (uv | 0x80000000u);
    atomicMax(&smaxk[n], key);
  }
}

// ---------------------------------------------------------------------------
// Pass 2: recompute mul2 tiles, E = exp(v - max); accumulate sum(E) and
// mul3 += E @ out4 via a second WMMA (E re-staged through LDS into A layout).
// ---------------------------------------------------------------------------
__global__ void k_softmax_mul3(const float* __restrict__ out2,
                               const float* __restrict__ out3,
                               const float* __restrict__ out4,
                               const unsigned* __restrict__ smaxk,
                               float* __restrict__ ssum, float* __restrict__ mul3) {
  __shared__ float Et[8 * 256];                        // one 16x16 tile per wave
  int tid  = threadIdx.x;
  int lane = tid & 31;
  int widx = tid >> 5;
  int half = lane >> 4;
  int lm   = lane & 15;
  int wg   = blockIdx.x * (blockDim.x >> 5) + widx;
  int n    = wg / (LT * MCH);
  int rem  = wg % (LT * MCH);
  int lt   = rem / MCH;
  int mc   = rem % MCH;
  int l0   = lt * 16;
  unsigned key = smaxk[n];
  float smx = (key & 0x80000000u) ? __uint_as_float(key & 0x7FFFFFFFu)
                                  : __uint_as_float(~key);
  float* E = Et + widx * 256;
  v8f dacc = {};
  float esum = 0.f;
  for (int mt = 0; mt < MT_PER_CH; ++mt) {
    int m0 = (mc * MT_PER_CH + mt) * 16;
    v8f c = mul2_tile(out2, out3, n, l0, m0, lane);
    float e[8];
#pragma unroll
    for (int r = 0; r < 8; ++r) { e[r] = __expf(c[r] - smx); esum += e[r]; }
    __syncthreads();                                   // uniform: same trip counts
#pragma unroll
    for (int r = 0; r < 8; ++r) E[(r + 8 * half) * 16 + lm] = e[r];  // D layout -> row major
    __syncthreads();
#pragma unroll
    for (int q = 0; q < 4; ++q) {                      // E(16x16) @ out4(16x9-pad-16)
      v2f a, b;
      a.x = E[lm * 16 + q * 4 + 2 * half];             // A row = lm, k = 2*half..+1
      a.y = E[lm * 16 + q * 4 + 2 * half + 1];
      float bx = 0.f, by = 0.f;
      if (lm < 9) {
        int mrow = m0 + q * 4 + 2 * half;
        bx = out4[((size_t)n * LL + mrow) * 9 + lm];
        by = out4[((size_t)n * LL + mrow + 1) * 9 + lm];
      }
      b.x = bx; b.y = by;
      dacc = __builtin_amdgcn_wmma_f32_16x16x4_f32(false, a, false, b,
                                                   (short)0, dacc, false, false);
    }
  }
#pragma unroll
  for (int off = 16; off > 0; off >>= 1) esum += __shfl_xor(esum, off, 32);
  if (lane == 0) atomicAdd(&ssum[n], esum);
  if (lm < 9) {
#pragma unroll
    for (int r = 0; r < 8; ++r) {
      int row = l0 + r + 8 * half;
      atomicAdd(&mul3[((size_t)n * LL + row) * 9 + lm], dacc[r]);
    }
  }
}

// ---------------------------------------------------------------------------
// mul4[n,c,j] = (sum_l coarse[n,c,l] * mul3u[n,l,j]) / ssum[n]
// ---------------------------------------------------------------------------
__global__ void k_mul4(const float* __restrict__ coarse, const float* __restrict__ mul3,
                       const float* __restrict__ ssum, float* __restrict__ mul4) {
  int idx = blockIdx.x * blockDim.x + threadIdx.x;
  if (idx >= NN * CC * 9) return;
  int n = idx / (CC * 9);
  int c = (idx / 9) % CC;
  int j = idx % 9;
  const float* cf = coarse + (size_t)(n * CC + c) * LL;
  const float* m3 = mul3 + (size_t)n * LL * 9 + j;
  float acc = 0.f;
  for (int l = 0; l < LL; ++l) acc = fmaf(cf[l], m3[(size_t)l * 9], acc);
  mul4[idx] = acc / ssum[n];
}

// ---------------------------------------------------------------------------
// mul5[n,c,j] = sum_m mul1[n,c,m] * mul4[n,m,j]
// ---------------------------------------------------------------------------
__global__ void k_mul5(const float* __restrict__ mul1, const float* __restrict__ mul4,
                       float* __restrict__ mul5) {
  int idx = blockIdx.x * blockDim.x + threadIdx.x;
  if (idx >= NN * CC * 9) return;
  int n = idx / (CC * 9);
  int c = (idx / 9) % CC;
  int j = idx % 9;
  const float* a = mul1 + (size_t)(n * CC + c) * CC;
  float acc = 0.f;
#pragma unroll 8
  for (int m = 0; m < CC; ++m) acc = fmaf(a[m], mul4[(n * CC + m) * 9 + j], acc);
  mul5[idx] = acc;
}

// ---------------------------------------------------------------------------
// Dynamic depthwise convs (d = 1,2,3) + BN batch statistics (wave-reduced)
// block = (n, c, 4-row strip); 256 threads = 4 rows x 64 cols
// ---------------------------------------------------------------------------
__global__ void k_dwconv(const float* __restrict__ x, const float* __restrict__ mul5,
                         const float* __restrict__ wd0, const float* __restrict__ wd1,
                         const float* __restrict__ wd2,
                         float* __restrict__ y0, float* __restrict__ y1,
                         float* __restrict__ y2,
                         float* __restrict__ bnsum, float* __restrict__ bnsq) {
  int b  = blockIdx.x;
  int hb = b % HB;
  int c  = (b / HB) % CC;
  int n  = b / (HB * CC);
  int tid = threadIdx.x;
  int w = tid & 63;
  int h = hb * 4 + (tid >> 6);
  const float* xc = x + (size_t)(n * CC + c) * LL;
  float k0[9], k1[9], k2[9];
#pragma unroll
  for (int t = 0; t < 9; ++t) {
    float m = mul5[(n * CC + c) * 9 + t];
    k0[t] = m * wd0[c * 9 + t];
    k1[t] = m * wd1[c * 9 + t];
    k2[t] = m * wd2[c * 9 + t];
  }
  float a0 = 0.f, a1 = 0.f, a2 = 0.f;
#pragma unroll
  for (int i = 0; i < 3; ++i) {
#pragma unroll
    for (int j = 0; j < 3; ++j) {
      int t = i * 3 + j;
      { int hh = h + (i - 1),     ww = w + (j - 1);
        if ((unsigned)hh < HH && (unsigned)ww < WW) a0 = fmaf(k0[t], xc[hh * WW + ww], a0); }
      { int hh = h + 2 * (i - 1), ww = w + 2 * (j - 1);
        if ((unsigned)hh < HH && (unsigned)ww < WW) a1 = fmaf(k1[t], xc[hh * WW + ww], a1); }
      { int hh = h + 3 * (i - 1), ww = w + 3 * (j - 1);
        if ((unsigned)hh < HH && (unsigned)ww < WW) a2 = fmaf(k2[t], xc[hh * WW + ww], a2); }
    }
  }
  size_t oidx = (size_t)(n * CC + c) * LL + h * WW + w;
  y0[oidx] = a0; y1[oidx] = a1; y2[oidx] = a2;
  float s0 = a0, q0 = a0 * a0, s1 = a1, q1 = a1 * a1, s2 = a2, q2 = a2 * a2;
#pragma unroll
  for (int off = 16; off > 0; off >>= 1) {
    s0 += __shfl_xor(s0, off, 32); q0 += __shfl_xor(q0, off, 32);
    s1 += __shfl_xor(s1, off, 32); q1 += __shfl_xor(q1, off, 32);
    s2 += __shfl_xor(s2, off, 32); q2 += __shfl_xor(q2, off, 32);
  }
  if ((tid & 31) == 0) {
    atomicAdd(&bnsum[0 * CC + c], s0); atomicAdd(&bnsq[0 * CC + c], q0);
    atomicAdd(&bnsum[1 * CC + c], s1); atomicAdd(&bnsq[1 * CC + c], q1);
    atomicAdd(&bnsum[2 * CC + c], s2); atomicAdd(&bnsq[2 * CC + c], q2);
  }
}

// ---------------------------------------------------------------------------
// Finalize BN: scale = g*rsqrt(var+eps), shift = be - mean*scale
// ---------------------------------------------------------------------------
__global__ void k_bnfin(const float* __restrict__ bnsum, const float* __restrict__ bnsq,
                        const float* __restrict__ g0, const float* __restrict__ be0,
                        const float* __restrict__ g1, const float* __restrict__ be1,
                        const float* __restrict__ g2, const float* __restrict__ be2,
                        float* __restrict__ bnscale, float* __restrict__ bnshift) {
  int t = blockIdx.x * blockDim.x + threadIdx.x;
  if (t >= 3 * CC) return;
  int d = t / CC, c = t % CC;
  const float cnt = (float)(NN * HH * WW);
  float mean = bnsum[t] / cnt;
  float var  = bnsq[t] / cnt - mean * mean;
  const float* g  = (d == 0) ? g0 : (d == 1) ? g1 : g2;
  const float* be = (d == 0) ? be0 : (d == 1) ? be1 : be2;
  float sc = g[c] * rsqrtf(var + 1e-5f);
  bnscale[t] = sc;
  bnshift[t] = be[c] - mean * sc;
}

// ---------------------------------------------------------------------------
// Final 1x1 conv over concat([x, relu(bn(y0)), relu(bn(y1)), relu(bn(y2))])
// ---------------------------------------------------------------------------
__global__ void k_final(const float* __restrict__ x,
                        const float* __restrict__ y0, const float* __restrict__ y1,
                        const float* __restrict__ y2,
                        const float* __restrict__ bnscale, const float* __restrict__ bnshift,
                        const float* __restrict__ w4, const float* __restrict__ b4,
                        float* __restrict__ out) {
  int idx = blockIdx.x * blockDim.x + threadIdx.x;     // N*C*L threads
  int n = idx / (CC * LL);
  int o = (idx / LL) % CC;
  int l = idx % LL;
  const float* wr = w4 + o * 4 * CC;
  float acc = b4[o];
  const float* xb = x + (size_t)n * CC * LL + l;
#pragma unroll 8
  for (int c = 0; c < CC; ++c) acc = fmaf(wr[c], xb[(size_t)c * LL], acc);
  const float* ys[3] = { y0, y1, y2 };
  for (int d = 0; d < 3; ++d) {
    const float* yb = ys[d] + (size_t)n * CC * LL + l;
    const float* sc = bnscale + d * CC;
    const float* sh = bnshift + d * CC;
    const float* wd = wr + (d + 1) * CC;
#pragma unroll 8
    for (int c = 0; c < CC; ++c) {
      float v = fmaf(yb[(size_t)c * LL], sc[c], sh[c]);
      v = fmaxf(v, 0.f);
      acc = fmaf(wd[c], v, acc);
    }
  }
  out[idx] = acc;
}

// ---------------------------------------------------------------------------
extern "C" void kernel_launch(void* const* d_in, const int* in_sizes, int n_in,
                              void* d_out, int out_size, void* d_ws, size_t ws_size,
                              hipStream_t stream) {
  const float* x      = (const float*)d_in[0];
  const float* coarse = (const float*)d_in[1];
  const float* w1  = (const float*)d_in[2];  const float* b1  = (const float*)d_in[3];
  const float* w2  = (const float*)d_in[4];  const float* b2  = (const float*)d_in[5];
  const float* w3  = (const float*)d_in[6];  const float* b3  = (const float*)d_in[7];
  const float* pcw1 = (const float*)d_in[8];  const float* pcb1 = (const float*)d_in[9];
  const float* pcw2 = (const float*)d_in[10]; const float* pcb2 = (const float*)d_in[11];
  const float* pcw3 = (const float*)d_in[12]; const float* pcb3 = (const float*)d_in[13];
  const float* wd0 = (const float*)d_in[14];
  const float* wd1 = (const float*)d_in[15];
  const float* wd2 = (const float*)d_in[16];
  const float* g0  = (const float*)d_in[17]; const float* be0 = (const float*)d_in[18];
  const float* g1  = (const float*)d_in[19]; const float* be1 = (const float*)d_in[20];
  const float* g2  = (const float*)d_in[21]; const float* be2 = (const float*)d_in[22];
  const float* w4  = (const float*)d_in[23]; const float* b4  = (const float*)d_in[24];
  float* out = (float*)d_out;

  float* ws = (float*)d_ws;
  size_t off = 0;
  float* out1  = ws + off; off += (size_t)NN * CC * LL;
  float* out2  = ws + off; off += (size_t)NN * CC * LL;
  float* out3  = ws + off; off += (size_t)NN * CC * LL;
  float* out4  = ws + off; off += (size_t)NN * LL * 9;
  float* mul3  = ws + off; off += (size_t)NN * LL * 9;
  float* mul1  = ws + off; off += (size_t)NN * CC * CC;
  float* mul4  = ws + off; off += (size_t)NN * CC * 9;
  float* mul5b = ws + off; off += (size_t)NN * CC * 9;
  float* y0    = ws + off; off += (size_t)NN * CC * LL;
  float* y1    = ws + off; off += (size_t)NN * CC * LL;
  float* y2    = ws + off; off += (size_t)NN * CC * LL;
  float* bnsum   = ws + off; off += 3 * CC;
  float* bnsq    = ws + off; off += 3 * CC;
  float* bnscale = ws + off; off += 3 * CC;
  float* bnshift = ws + off; off += 3 * CC;
  unsigned* smaxk = (unsigned*)(ws + off); off += NN;
  float* ssum  = ws + off; off += NN;

  k_init<<<(NN * LL * 9 + 255) / 256, 256, 0, stream>>>(mul3, ssum, smaxk, bnsum, bnsq);

  k_out123<<<(3 * NN * CC * LL) / 256, 256, 0, stream>>>(
      x, w1, b1, w2, b2, w3, b3, out1, out2, out3);

  k_paired<<<(NN * LL) / 256, 256, 0, stream>>>(
      x, pcw1, pcb1, pcw2, pcb2, pcw3, pcb3, out4);

  k_mul1<<<(NN * CC * CC) / 256, 256, 0, stream>>>(out1, coarse, mul1);

  const int smblocks = (NN * LT * MCH) / 8;            // 8 waves / block
  k_softmax_max<<<smblocks, 256, 0, stream>>>(out2, out3, smaxk);
  k_softmax_mul3<<<smblocks, 256, 0, stream>>>(out2, out3, out4, smaxk, ssum, mul3);

  k_mul4<<<(NN * CC * 9 + 255) / 256, 256, 0, stream>>>(coarse, mul3, ssum, mul4);
  k_mul5<<<(NN * CC * 9 + 255) / 256, 256, 0, stream>>>(mul1, mul4, mul5b);

  k_dwconv<<<NN * CC * HB, 256, 0, stream>>>(
      x, mul5b, wd0, wd1, wd2, y0, y1, y2, bnsum, bnsq);

  k_bnfin<<<1, 256, 0, stream>>>(bnsum, bnsq, g0, be0, g1, be1, g2, be2,
                                 bnscale, bnshift);

  k_final<<<(NN * CC * LL) / 256, 256, 0, stream>>>(
      x, y0, y1, y2, bnscale, bnshift, w4, b4, out);
}